// LogEntmax15_21096879358426
// MI455X (gfx1250) — compile-verified
//
#include <hip/hip_runtime.h>
#include <math.h>
#include <stdint.h>

// log(entmax15(x)) over last axis. x: (1,16,2048,2048) fp32 -> same-shape fp32.
// Strategy: one wave32 per row (2048 elems = 64 regs/lane).
//   1. async-copy row global->LDS (global_load_async_to_lds_b128, ASYNCcnt)
//   2. ds_load_b128 into registers, wave-reduce max (shfl_xor, wave32)
//   3. bisection on f(tau)=sum(max(y-tau,0)^2)=1  (26 iters, register resident)
//   4. two closed-form refinements on the induced support (matches reference
//      tau = mean - sqrt((1-ss)/k) exactly on the converged support)
//   5. write log(max(y-tau,0)^2) or -inf, 128-bit coalesced stores.

#define NCOL 2048
#define ROWS_PER_BLOCK 8          // 8 waves * 32 lanes = 256 threads
#define EPL 64                    // elements per lane = NCOL/32
#define F4PL 16                   // float4 chunks per lane

__device__ __forceinline__ float waveAdd(float v) {
#pragma unroll
  for (int d = 16; d > 0; d >>= 1) v += __shfl_xor(v, d, 32);
  return v;
}

__device__ __forceinline__ float waveMax(float v) {
#pragma unroll
  for (int d = 16; d > 0; d >>= 1) v = fmaxf(v, __shfl_xor(v, d, 32));
  return v;
}

__global__ __launch_bounds__(256) void LogEntmax15_kernel(
    const float* __restrict__ x, float* __restrict__ out, int nrows) {
  __shared__ float lds[ROWS_PER_BLOCK * NCOL];  // 64 KB

  const int lane = threadIdx.x & 31;
  const int wave = threadIdx.x >> 5;
  const int row  = blockIdx.x * ROWS_PER_BLOCK + wave;
  if (row >= nrows) return;  // wave-uniform

  const float* __restrict__ grow = x + (size_t)row * NCOL;
  float* __restrict__ orow = out + (size_t)row * NCOL;
  float* lrow = &lds[wave * NCOL];
  // LDS aperture: low 32 bits of a flat LDS address are the LDS byte offset.
  const uint32_t lbase = (uint32_t)(uintptr_t)lrow;

  // ---- 1) async copy row to LDS: 16 x b128 per wave (512B/instr) ----------
#pragma unroll
  for (int i = 0; i < F4PL; ++i) {
    uint32_t loff = lbase + (uint32_t)((i * 32 + lane) * 16);
    const void* gp = (const void*)(grow + (i * 32 + lane) * 4);
    asm volatile("global_load_async_to_lds_b128 %0, %1, off"
                 :: "v"(loff), "v"(gp) : "memory");
  }
  asm volatile("s_wait_asynccnt 0" ::: "memory");

  // ---- 2) LDS -> registers, then wave max ---------------------------------
  float y[EPL];
  {
    const float4* l4 = (const float4*)lrow;
#pragma unroll
    for (int i = 0; i < F4PL; ++i) {
      float4 v = l4[i * 32 + lane];
      y[4 * i + 0] = v.x; y[4 * i + 1] = v.y;
      y[4 * i + 2] = v.z; y[4 * i + 3] = v.w;
    }
  }
  float m = -__builtin_inff();
#pragma unroll
  for (int j = 0; j < EPL; ++j) m = fmaxf(m, y[j]);
  m = waveMax(m);
  // y = (x - max)/2  (== x/2 - max(x/2)); now max(y) == 0, tau* in [-1, 0)
#pragma unroll
  for (int j = 0; j < EPL; ++j) y[j] = (y[j] - m) * 0.5f;

  // ---- 3) bisection on f(tau) = sum max(y-tau,0)^2 - 1 --------------------
  float lo = -1.0f, hi = 0.0f;
#pragma unroll 1
  for (int it = 0; it < 26; ++it) {
    float tau = 0.5f * (lo + hi);
    float s = 0.0f;
#pragma unroll
    for (int j = 0; j < EPL; ++j) {
      float t = y[j] - tau;
      t = t > 0.0f ? t : 0.0f;
      s = fmaf(t, t, s);
    }
    s = waveAdd(s);
    if (s >= 1.0f) lo = tau; else hi = tau;
  }

  // ---- 4) closed-form refinement on induced support (2 rounds) ------------
  float tau = lo;
#pragma unroll 1
  for (int r = 0; r < 2; ++r) {
    float k = 0.0f, s1 = 0.0f, s2 = 0.0f;
#pragma unroll
    for (int j = 0; j < EPL; ++j) {
      if (y[j] > tau) {
        k += 1.0f;
        s1 += y[j];
        s2 = fmaf(y[j], y[j], s2);
      }
    }
    k = waveAdd(k);
    s1 = waveAdd(s1);
    s2 = waveAdd(s2);
    float mean  = s1 / k;                 // k >= 1 always (max elem in support)
    float ss    = fmaf(-mean, s1, s2);    // k * (mean_sq - mean^2)
    float delta = (1.0f - ss) / k;
    delta = fmaxf(delta, 0.0f);
    tau = mean - __builtin_sqrtf(delta);
  }

  // ---- 5) p = max(y-tau,0)^2 ; out = p>0 ? log(p) : -inf ------------------
  const float ninf = -__builtin_inff();
  float4* o4 = (float4*)orow;
#pragma unroll
  for (int i = 0; i < F4PL; ++i) {
    float4 v;
    float t0 = y[4 * i + 0] - tau;
    float t1 = y[4 * i + 1] - tau;
    float t2 = y[4 * i + 2] - tau;
    float t3 = y[4 * i + 3] - tau;
    v.x = (t0 > 0.0f) ? __logf(t0 * t0) : ninf;
    v.y = (t1 > 0.0f) ? __logf(t1 * t1) : ninf;
    v.z = (t2 > 0.0f) ? __logf(t2 * t2) : ninf;
    v.w = (t3 > 0.0f) ? __logf(t3 * t3) : ninf;
    o4[i * 32 + lane] = v;
  }
}

extern "C" void kernel_launch(void* const* d_in, const int* in_sizes, int n_in,
                              void* d_out, int out_size, void* d_ws, size_t ws_size,
                              hipStream_t stream) {
  const float* x = (const float*)d_in[0];
  float* out = (float*)d_out;
  const int total = in_sizes[0];        // 1*16*2048*2048
  const int nrows = total / NCOL;       // 32768
  dim3 grid((nrows + ROWS_PER_BLOCK - 1) / ROWS_PER_BLOCK);
  LogEntmax15_kernel<<<grid, 256, 0, stream>>>(x, out, nrows);
}